// EinsumGateUp_62878321214308
// MI455X (gfx1250) — compile-verified
//
#include <hip/hip_runtime.h>

// Problem constants (A=1 folded away)
#define Hd   1024   // hidden dim
#define Nd   1024   // weight inner dim (summed)
#define Ed   8      // experts
#define Bd   32     // batch
#define Md   32     // rows (summed)

#define WBLOCKS ((Ed * Hd) / 8)   // 1024 blocks for the weight stream

typedef __attribute__((ext_vector_type(2))) float v2f;
typedef __attribute__((ext_vector_type(8))) float v8f;

// ---------------------------------------------------------------------------
// Fused streaming reductions (both independent, both pure-bandwidth; fusing
// them into one grid lets them share the HBM-saturated window instead of
// serializing a 4MB tail after the 32MB stream).
//
//  blocks [0, 1024):    wsum[e*Hd + h] = sum_n weight[e][h][n]
//                       one wave32 per (e,h) row; 8 x b128 loads per lane
//                       (512B/wave/iter, fully coalesced) + xor-tree reduce.
//  blocks [1024, 1056): hsum[b*Hd + h] = sum_m hidden[b][m][h]
//                       one block per b; thread t owns float4 strip h=[4t,4t+4);
//                       each m-iteration is one contiguous 4KB blockwide load.
// ---------------------------------------------------------------------------
__global__ __launch_bounds__(256) void reduce_fused_kernel(
    const float* __restrict__ w, const float* __restrict__ hid,
    float* __restrict__ wsum, float* __restrict__ hsum) {
  if (blockIdx.x < WBLOCKS) {
    // ---- weight row-sum path ----
    const int lane = threadIdx.x & 31;
    const int wave = threadIdx.x >> 5;
    const int row  = blockIdx.x * 8 + wave;          // [0, Ed*Hd)
    const float4* wr = (const float4*)(w + (size_t)row * Nd);
    float s = 0.f;
#pragma unroll
    for (int i = 0; i < Nd / (32 * 4); ++i) {        // 8 iterations
      float4 v = wr[i * 32 + lane];
      s += (v.x + v.y) + (v.z + v.w);
    }
#pragma unroll
    for (int off = 16; off > 0; off >>= 1)
      s += __shfl_xor(s, off, 32);
    if (lane == 0) wsum[row] = s;
  } else {
    // ---- hidden column-sum path ----
    const int b = blockIdx.x - WBLOCKS;              // [0, Bd)
    const int t = threadIdx.x;                       // 0..255, Hd/4 = 256
    const float4* base = (const float4*)(hid + (size_t)b * Md * Hd);
    float4 acc = {0.f, 0.f, 0.f, 0.f};
#pragma unroll 8
    for (int m = 0; m < Md; ++m) {
      float4 v = base[m * (Hd / 4) + t];
      acc.x += v.x; acc.y += v.y; acc.z += v.z; acc.w += v.w;
    }
    ((float4*)hsum)[b * (Hd / 4) + t] = acc;
  }
}

// ---------------------------------------------------------------------------
// Finisher: out[b][e] = mask[b][e] * sum_h hsum[b][h] * wsum[e][h]
// (32 x 1024) x (1024 x 8) via V_WMMA_F32_16X16X4_F32, K in steps of 4.
// 2 waves: wave w computes output rows [16w, 16w+16), cols 0..15 (E=8 valid,
// cols 8..15 fed zeros branchlessly so EXEC stays all-ones for WMMA).
//
// A (16x4 f32) lane layout (ISA 7.12.2): lanes 0-15 hold K=0(.x),1(.y);
// lanes 16-31 hold K=2(.x),3(.y), M = lane%16. B (4x16) mirrored, N=lane%16.
// C/D (16x16 f32): VGPR v -> row M = v + 8*(lane>=16), col N = lane%16.
// ---------------------------------------------------------------------------
__global__ __launch_bounds__(64) void gemm_wmma_kernel(
    const float* __restrict__ hsum, const float* __restrict__ wsum,
    const float* __restrict__ mask, float* __restrict__ out) {
  const int lane    = threadIdx.x & 31;
  const int wave    = threadIdx.x >> 5;     // 0 or 1
  const int rowbase = wave * 16;
  const int l       = lane & 15;
  const int hi      = (lane >> 4) & 1;      // K sub-pair select

  const float* arow   = hsum + (size_t)(rowbase + l) * Hd;
  const float  bvalid = (l < Ed) ? 1.f : 0.f;                 // zero pad cols 8..15
  const float* brow   = wsum + (size_t)(l < Ed ? l : 0) * Hd; // branchless, no null

  v8f c = {};
  for (int k0 = 0; k0 < Hd; k0 += 4) {
    const int ka = k0 + hi * 2;
    v2f a, b;
    a.x = arow[ka];
    a.y = arow[ka + 1];
    b.x = brow[ka] * bvalid;
    b.y = brow[ka + 1] * bvalid;
    // 8 args: (neg_a, A, neg_b, B, c_mod, C, reuse_a, reuse_b)
    c = __builtin_amdgcn_wmma_f32_16x16x4_f32(
        false, a, false, b, (short)0, c, false, false);
  }

#pragma unroll
  for (int v = 0; v < 8; ++v) {
    const int b = rowbase + v + hi * 8;     // output batch row
    const int e = l;                        // expert col
    if (e < Ed) out[b * Ed + e] = c[v] * mask[b * Ed + e];
  }
}

// ---------------------------------------------------------------------------
extern "C" void kernel_launch(void* const* d_in, const int* in_sizes, int n_in,
                              void* d_out, int out_size, void* d_ws, size_t ws_size,
                              hipStream_t stream) {
  const float* hidden   = (const float*)d_in[0];  // (1,32,32,1024) f32
  const float* sparsity = (const float*)d_in[1];  // (1,32,1,8)    f32
  const float* weight   = (const float*)d_in[2];  // (1,8,1024,1024) f32
  float*       out      = (float*)d_out;          // (1,32,8) f32

  float* wsum = (float*)d_ws;                     // Ed*Hd   =  8192 f32 (32KB)
  float* hsum = wsum + Ed * Hd;                   // Bd*Hd   = 32768 f32 (128KB)

  reduce_fused_kernel<<<WBLOCKS + Bd, 256, 0, stream>>>(weight, hidden, wsum, hsum);
  gemm_wmma_kernel<<<1, 64, 0, stream>>>(hsum, wsum, sparsity, out);
}